// GramEmbedding_60095182406250
// MI455X (gfx1250) — compile-verified
//
#include <hip/hip_runtime.h>
#include <hip/hip_bf16.h>

#define F_DIM 1024
#define H_DIM 256
#define N_DIM 1280   // F + H
#define E_DIM 128
#define A_DIM 128
#define B_DIM 256

typedef __attribute__((ext_vector_type(2))) float v2f;
typedef __attribute__((ext_vector_type(8))) float v8f;

__device__ __forceinline__ v8f wmma_f32_16x16x4(v2f a, v2f b, v8f c) {
#if defined(__HIP_DEVICE_COMPILE__) && \
    __has_builtin(__builtin_amdgcn_wmma_f32_16x16x4_f32)
  // (neg_a, A, neg_b, B, c_mod, C, reuse_a, reuse_b)
  return __builtin_amdgcn_wmma_f32_16x16x4_f32(false, a, false, b, (short)0, c,
                                               false, false);
#else
  // Host pass (builtins unavailable) or builtin missing on device:
  // never executed on host; if taken on device, histogram shows wmma=0.
  return c;
#endif
}

// -------- Kernel 1: q = feat @ W1 (1024x128), k = full @ W2 (1280x128) ------
// One wave per 16x16 output tile; K = 128 in steps of 4 (fp32 WMMA).
__global__ void __launch_bounds__(32)
qk_kernel(const float* __restrict__ feat, const float* __restrict__ hid,
          const float* __restrict__ W1, const float* __restrict__ W2,
          float* __restrict__ qbuf, float* __restrict__ kbuf) {
  const int lane = threadIdx.x;
  const int nl = lane & 15;
  const int hi = lane >> 4;          // 0 or 1 -> K sub-pair / row offset
  const int n0 = blockIdx.x * 16;    // output column tile (A dim, 8 tiles)
  const int mt = blockIdx.y;         // 0..143 row tiles (64 q + 80 k)

  const float* W;
  const float* src;
  float* dst;
  int srcrow0, dstrow0;
  if (mt < 64) {                     // q part
    W = W1; src = feat; dst = qbuf;
    srcrow0 = mt * 16; dstrow0 = mt * 16;
  } else {                           // k part (full = [feat ; hid])
    W = W2; dst = kbuf;
    const int r0 = (mt - 64) * 16;   // 16-row tiles never straddle 1024
    dstrow0 = r0;
    if (r0 < F_DIM) { src = feat; srcrow0 = r0; }
    else            { src = hid;  srcrow0 = r0 - F_DIM; }
  }

  const int am = srcrow0 + nl;       // A-matrix row held by this lane
  v8f c = {};
  for (int k0 = 0; k0 < E_DIM; k0 += 4) {
    const int kb = k0 + hi * 2;
    v2f a = *(const v2f*)(src + am * E_DIM + kb);
    v2f b;
    b.x = W[(kb + 0) * A_DIM + n0 + nl];
    b.y = W[(kb + 1) * A_DIM + n0 + nl];
    c = wmma_f32_16x16x4(a, b, c);
  }
#pragma unroll
  for (int v = 0; v < 8; ++v) {
    const int row = dstrow0 + v + 8 * hi;
    dst[row * A_DIM + n0 + nl] = c[v];
  }
}

// -------- Kernel 2: escore[f,n] = mask ? exp(sum_a u_a tanh(q+k+b)) : 0 -----
// Dominant cost: 168M tanh -> TRANS pipe. LDS tiles padded to 132 floats
// so ks[tx*132+a] hits bank (4*tx+a)%64 -> conflict-free across a wave.
__global__ void __launch_bounds__(256)
score_kernel(const float* __restrict__ qbuf, const float* __restrict__ kbuf,
             const float* __restrict__ bvec, const float* __restrict__ uvec,
             const unsigned char* __restrict__ mask,
             float* __restrict__ escore) {
  __shared__ float qs[16 * 132];
  __shared__ float ks[16 * 132];
  __shared__ float ub[128];
  __shared__ float bb[128];
  const int tid = threadIdx.x;
  const int n0 = blockIdx.x * 16;
  const int f0 = blockIdx.y * 16;

  for (int i = tid; i < 16 * 128; i += 256) {
    const int r = i >> 7, cc = i & 127;
    qs[r * 132 + cc] = qbuf[(f0 + r) * A_DIM + cc];
    ks[r * 132 + cc] = kbuf[(n0 + r) * A_DIM + cc];
  }
  if (tid < 128) { ub[tid] = uvec[tid]; bb[tid] = bvec[tid]; }
  __syncthreads();

  const int tx = tid & 15;   // n within tile
  const int ty = tid >> 4;   // f within tile
  const float* qrow = &qs[ty * 132];
  const float* krow = &ks[tx * 132];
  float s = 0.f;
#pragma unroll 4
  for (int a = 0; a < A_DIM; ++a)
    s += ub[a] * tanhf(qrow[a] + krow[a] + bb[a]);

  const int f = f0 + ty, n = n0 + tx;
  const float e = mask[f * N_DIM + n] ? __expf(s) : 0.f;
  escore[f * N_DIM + n] = e;
}

// -------- Kernel 3: rowsum[f] = sum_n escore[f,n] (guard zero -> 1) ---------
__global__ void __launch_bounds__(256)
rowsum_kernel(const float* __restrict__ escore, float* __restrict__ rowsum) {
  __shared__ float red[256];
  const int f = blockIdx.x;
  const int tid = threadIdx.x;
  float s = 0.f;
  for (int n = tid; n < N_DIM; n += 256) s += escore[f * N_DIM + n];
  red[tid] = s;
  __syncthreads();
  for (int off = 128; off > 0; off >>= 1) {
    if (tid < off) red[tid] += red[tid + off];
    __syncthreads();
  }
  if (tid == 0) {
    const float v = red[0];
    rowsum[f] = (v == 0.f) ? 1.f : v;
  }
}

__device__ __forceinline__ float full_at(const float* feat, const float* hid,
                                         int n, int e) {
  return (n < F_DIM) ? feat[n * E_DIM + e] : hid[(n - F_DIM) * E_DIM + e];
}

// -------- Kernel 4: context = (escore @ full) / rowsum (1024x128, K=1280) ---
__global__ void __launch_bounds__(32)
context_kernel(const float* __restrict__ escore, const float* __restrict__ feat,
               const float* __restrict__ hid, const float* __restrict__ rowsum,
               float* __restrict__ context) {
  const int lane = threadIdx.x;
  const int nl = lane & 15, hi = lane >> 4;
  const int n0 = blockIdx.x * 16;   // E tile (8)
  const int m0 = blockIdx.y * 16;   // F tile (64)
  const int am = m0 + nl;
  v8f c = {};
  for (int k0 = 0; k0 < N_DIM; k0 += 4) {
    const int kb = k0 + hi * 2;
    v2f a = *(const v2f*)(escore + am * N_DIM + kb);
    v2f b;
    b.x = full_at(feat, hid, kb + 0, n0 + nl);
    b.y = full_at(feat, hid, kb + 1, n0 + nl);
    c = wmma_f32_16x16x4(a, b, c);
  }
#pragma unroll
  for (int v = 0; v < 8; ++v) {
    const int row = m0 + v + 8 * hi;
    context[row * E_DIM + n0 + nl] = c[v] / rowsum[row];
  }
}

// -------- Kernel 5: out = values @ context (256x128, K=1024) ----------------
__global__ void __launch_bounds__(32)
out_kernel(const float* __restrict__ values, const float* __restrict__ context,
           float* __restrict__ out) {
  const int lane = threadIdx.x;
  const int nl = lane & 15, hi = lane >> 4;
  const int n0 = blockIdx.x * 16;   // E tile (8)
  const int m0 = blockIdx.y * 16;   // B tile (16)
  const int am = m0 + nl;
  v8f c = {};
  for (int k0 = 0; k0 < F_DIM; k0 += 4) {
    const int kb = k0 + hi * 2;
    v2f a = *(const v2f*)(values + am * F_DIM + kb);
    v2f b;
    b.x = context[(kb + 0) * E_DIM + n0 + nl];
    b.y = context[(kb + 1) * E_DIM + n0 + nl];
    c = wmma_f32_16x16x4(a, b, c);
  }
#pragma unroll
  for (int v = 0; v < 8; ++v)
    out[(m0 + v + 8 * hi) * E_DIM + n0 + nl] = c[v];
}

extern "C" void kernel_launch(void* const* d_in, const int* in_sizes, int n_in,
                              void* d_out, int out_size, void* d_ws,
                              size_t ws_size, hipStream_t stream) {
  (void)in_sizes; (void)n_in; (void)out_size; (void)ws_size;
  const float* values = (const float*)d_in[0];        // [256,1024]
  const float* feat   = (const float*)d_in[1];        // [1024,128]
  const float* hid    = (const float*)d_in[2];        // [256,128]
  const float* W1     = (const float*)d_in[3];        // [128,128]
  const float* W2     = (const float*)d_in[4];        // [128,128]
  const float* bvec   = (const float*)d_in[5];        // [128]
  const float* uvec   = (const float*)d_in[6];        // [128]
  const unsigned char* mask = (const unsigned char*)d_in[7]; // [1024,1280] bool

  float* ws = (float*)d_ws;
  float* qbuf    = ws;                          // 1024*128
  float* kbuf    = qbuf + F_DIM * A_DIM;        // 1280*128
  float* escore  = kbuf + N_DIM * A_DIM;        // 1024*1280
  float* rowsum  = escore + F_DIM * N_DIM;      // 1024
  float* context = rowsum + F_DIM;              // 1024*128  (~7 MB total)

  qk_kernel<<<dim3(8, 144), 32, 0, stream>>>(feat, hid, W1, W2, qbuf, kbuf);
  score_kernel<<<dim3(N_DIM / 16, F_DIM / 16), 256, 0, stream>>>(
      qbuf, kbuf, bvec, uvec, mask, escore);
  rowsum_kernel<<<F_DIM, 256, 0, stream>>>(escore, rowsum);
  context_kernel<<<dim3(8, F_DIM / 16), 32, 0, stream>>>(escore, feat, hid,
                                                         rowsum, context);
  out_kernel<<<dim3(8, B_DIM / 16), 32, 0, stream>>>(values, context,
                                                     (float*)d_out);
}